// QC_67473936220747
// MI455X (gfx1250) — compile-verified
//
#include <hip/hip_runtime.h>

typedef float v2f __attribute__((ext_vector_type(2)));
typedef float v8f __attribute__((ext_vector_type(8)));

#define NGATES 8
#define NTILES 256
#define SDIM   64
#define BDIM   1024
#define TILE   (SDIM * BDIM)   // 65536

// One gate: out[t, base[i,c]] = sum_j U[i,j] * in[perm[t]*TILE + base[j,c]]
// Grid: (NTILES, 8 column-chunks). Block: 256 threads = 8 waves.
// Each wave handles one 16-column N-tile of the 64x64 x 64x1024 fp32 GEMM
// using V_WMMA_F32_16X16X4_F32 (4 M-tiles x 16 K-steps).
__global__ __launch_bounds__(256)
void QC_gate_kernel(const float* __restrict__ in,
                    float* __restrict__ out,
                    const float* __restrict__ U,      // [64][64] this gate
                    const int* __restrict__ perm,     // [NTILES] this gate
                    const int* __restrict__ base)     // [64][1024]
{
    __shared__ float ldsU[SDIM * SDIM];               // 16 KB

    const int t     = blockIdx.x;
    const int chunk = blockIdx.y;
    const int tid   = threadIdx.x;

    // Stage U into LDS: 4096 floats = 1024 float4, 256 threads x 4
    {
        const float4* src = (const float4*)U;
        float4*       dst = (float4*)ldsU;
#pragma unroll
        for (int r = 0; r < 4; ++r)
            dst[tid + r * 256] = src[tid + r * 256];
    }
    __syncthreads();

    const int w  = tid >> 5;          // wave id 0..7
    const int ln = tid & 15;          // lane % 16 -> column within N-tile
    const int hh = (tid >> 4) & 1;    // lane half (0: lanes 0-15, 1: lanes 16-31)
    const int c  = chunk * 128 + w * 16 + ln;   // global column 0..1023

    const float* __restrict__ inP  = in  + (long)perm[t] * TILE;
    float*       __restrict__ outP = out + (long)t       * TILE;

    // ---- Gather all 16 B fragments (K=4 each): lane holds rows
    //      kk*4 + 2*hh + {0,1} at column c (ISA 32-bit B 4x16 layout) ----
    v2f bf[16];
#pragma unroll
    for (int kk = 0; kk < 16; ++kk) {
        const int j0 = kk * 4 + hh * 2;
        const int i0 = base[j0 * BDIM + c];
        const int i1 = base[(j0 + 1) * BDIM + c];
        bf[kk].x = inP[i0];
        bf[kk].y = inP[i1];
    }

    // ---- 4 M-tiles: 16 chained fp32 WMMAs each, then scatter D ----
#pragma unroll
    for (int m = 0; m < 4; ++m) {
        v8f acc = {};
#pragma unroll
        for (int kk = 0; kk < 16; ++kk) {
            // A 16x4 layout: lane holds U[m*16+ln][kk*4 + 2*hh + {0,1}]
            const v2f af =
                *(const v2f*)&ldsU[(m * 16 + ln) * SDIM + kk * 4 + hh * 2];
            acc = __builtin_amdgcn_wmma_f32_16x16x4_f32(
                /*neg_a=*/false, af, /*neg_b=*/false, bf[kk],
                /*c_mod=*/(short)0, acc, /*reuse_a=*/false, /*reuse_b=*/false);
        }
        // D 16x16 layout: VGPR v -> row m*16 + 8*hh + v, column c
#pragma unroll
        for (int v = 0; v < 8; ++v) {
            const int i = m * 16 + hh * 8 + v;
            outP[base[i * BDIM + c]] = acc[v];
        }
    }
}

extern "C" void kernel_launch(void* const* d_in, const int* in_sizes, int n_in,
                              void* d_out, int out_size, void* d_ws, size_t ws_size,
                              hipStream_t stream) {
    (void)in_sizes; (void)n_in; (void)out_size; (void)ws_size;

    const float* state = (const float*)d_in[0];   // [256][65536] f32
    const float* unis  = (const float*)d_in[1];   // [8][64][64]  f32
    const int*   idcs  = (const int*)  d_in[2];   // [8][256]     i32
    const int*   base  = (const int*)  d_in[3];   // [64][1024]   i32

    float* out = (float*)d_out;                   // 64 MB
    float* ws  = (float*)d_ws;                    // >= 64 MB scratch

    dim3 grid(NTILES, 8);
    dim3 block(256);

    // Ping-pong: even gates -> ws, odd gates -> out; gate 7 (odd) ends in d_out.
    const float* cur = state;
    for (int g = 0; g < NGATES; ++g) {
        float* dst = (g & 1) ? out : ws;
        QC_gate_kernel<<<grid, block, 0, stream>>>(
            cur, dst, unis + g * SDIM * SDIM, idcs + g * NTILES, base);
        cur = dst;
    }
}